// LSTM_46316927320185
// MI455X (gfx1250) — compile-verified
//
#include <hip/hip_runtime.h>

// Sizes from the reference
constexpr int kB    = 4096;   // batch
constexpr int kT    = 512;    // time steps
constexpr int kEMB  = 32;
constexpr int kHID  = 16;
constexpr int kG4   = 64;     // 4 * HID (i,f,g,o)

typedef float v2f __attribute__((ext_vector_type(2)));
typedef float v8f __attribute__((ext_vector_type(8)));

__device__ __forceinline__ float sigm(float x) {
    return 1.0f / (1.0f + __expf(-x));
}
__device__ __forceinline__ float tanh_fast(float x) {
    x = fminf(fmaxf(x, -15.0f), 15.0f);
    float e = __expf(2.0f * x);
    return (e - 1.0f) / (e + 1.0f);
}

// -------------------------------------------------------------------------
// Kernel A: x_proj_T[t][n][b] = sum_k w_ih[n][k]*embed[x[b][t]][k] + bias[n]
// One wave32 per (16-batch-row tile, 8 timesteps).
// WMMA form: D(16n x 16b) = A(w_ih, 16x4) x B(emb^T, 4x16), K=32 via 8 chunks.
// K-chunk outer / gate-tile inner: consecutive WMMAs hit independent
// accumulators, hiding the XDL RAW latency.
// -------------------------------------------------------------------------
__global__ void lstm_xproj(const int* __restrict__ x,
                           const float* __restrict__ embed,
                           const float* __restrict__ w_ih,
                           const float* __restrict__ b_ih,
                           const float* __restrict__ b_hh,
                           float* __restrict__ xp) {
    const int lane = threadIdx.x;
    const int m  = lane & 15;       // batch row within tile / n-row for A
    const int hi = lane >> 4;       // half-wave select (K pairing per ISA layout)
    const int b0 = blockIdx.x * 16;
    const int t0 = blockIdx.y * 8;

    // A = w_ih in 16x4-f32 A layout: lane<16 holds K={4c,4c+1}, lane>=16 K={4c+2,4c+3}
    v2f a[4][8];
#pragma unroll
    for (int g = 0; g < 4; ++g) {
        const float* wr = w_ih + (16 * g + m) * kEMB;
#pragma unroll
        for (int c = 0; c < 8; ++c) {
            a[g][c].x = wr[4 * c + 2 * hi];
            a[g][c].y = wr[4 * c + 2 * hi + 1];
        }
    }
    // bias for D element (g, r): n = 16g + 8*hi + r
    float bias[4][8];
#pragma unroll
    for (int g = 0; g < 4; ++g) {
        const float* bi = b_ih + 16 * g + 8 * hi;
        const float* bh = b_hh + 16 * g + 8 * hi;
#pragma unroll
        for (int r = 0; r < 8; ++r) bias[g][r] = bi[r] + bh[r];
    }

    for (int it = 0; it < 8; ++it) {
        const int t = t0 + it;
        // Gather: lanes (m, 0) and (m, 1) share the row index (L2-resident table)
        const int idx = x[(b0 + m) * kT + t];
        const float* er = embed + (long)idx * kEMB;
        v2f bm[8];
#pragma unroll
        for (int c = 0; c < 8; ++c) {
            bm[c].x = er[4 * c + 2 * hi];
            bm[c].y = er[4 * c + 2 * hi + 1];
        }
        v8f acc[4];
#pragma unroll
        for (int g = 0; g < 4; ++g)
#pragma unroll
            for (int r = 0; r < 8; ++r) acc[g][r] = bias[g][r];
        // chunk-outer, tile-inner: 4 independent WMMAs between dependent pairs
#pragma unroll
        for (int c = 0; c < 8; ++c)
#pragma unroll
            for (int g = 0; g < 4; ++g)
                acc[g] = __builtin_amdgcn_wmma_f32_16x16x4_f32(
                    false, a[g][c], false, bm[c], (short)0, acc[g], false, false);

        // Store D: element (g,r) at lane (m,hi) -> xp[(t*64 + 16g + 8hi + r)*B + b0 + m]
#pragma unroll
        for (int g = 0; g < 4; ++g) {
            float* dst = xp + ((long)t * kG4 + 16 * g + 8 * hi) * kB + b0 + m;
#pragma unroll
            for (int r = 0; r < 8; ++r) dst[(long)r * kB] = acc[g][r];
        }
    }
}

// -------------------------------------------------------------------------
// Kernel B: serial scan. One wave32 per 16 batch rows, loops T=512.
// gates_T(16n x 16b) = w_hh(A) x h^T(B) + xp_T(C), K=16 via 4 chunks.
// h state lives in the D layout (lane = batch row, reg = hidden idx), converted
// to the B layout each step with one half-wave shfl_xor exchange.
// -------------------------------------------------------------------------
__global__ void lstm_scan(const float* __restrict__ xp,
                          const float* __restrict__ w_hh,
                          const float* __restrict__ fc_w,
                          const float* __restrict__ fc_b,
                          float* __restrict__ out) {
    const int lane = threadIdx.x;
    const int m  = lane & 15;
    const int hi = lane >> 4;
    const int b0 = blockIdx.x * 16;

    // A = w_hh, static in registers
    v2f a[4][4];
#pragma unroll
    for (int g = 0; g < 4; ++g) {
        const float* wr = w_hh + (16 * g + m) * kHID;
#pragma unroll
        for (int c = 0; c < 4; ++c) {
            a[g][c].x = wr[4 * c + 2 * hi];
            a[g][c].y = wr[4 * c + 2 * hi + 1];
        }
    }

    float cst[8], h[8];
#pragma unroll
    for (int r = 0; r < 8; ++r) { cst[r] = 0.0f; h[r] = 0.0f; }

    for (int t = 0; t < kT; ++t) {
        // C = xp_T tile for this step (2x64B contiguous per load instruction)
        v8f acc[4];
        const float* base = xp + (long)t * kG4 * kB;
#pragma unroll
        for (int g = 0; g < 4; ++g) {
            const float* src = base + (long)(16 * g + 8 * hi) * kB + b0 + m;
#pragma unroll
            for (int r = 0; r < 8; ++r) acc[g][r] = src[(long)r * kB];
        }
        // Prefetch next step's slab (rows n=lane and n=lane+32 at our batch cols)
        if (t + 1 < kT) {
            const float* pf = xp + ((long)(t + 1) * kG4 + lane) * kB + b0;
            __builtin_prefetch(pf, 0, 1);
            __builtin_prefetch(pf + 32L * kB, 0, 1);
        }

        // Build B (h^T) from D-layout h: swap half-wave K groups
        float p[8];
#pragma unroll
        for (int r = 0; r < 8; ++r) p[r] = __shfl_xor(h[r], 16, 32);
        v2f bm[4];
        bm[0].x = hi ? p[2] : h[0];  bm[0].y = hi ? p[3] : h[1];
        bm[1].x = hi ? p[6] : h[4];  bm[1].y = hi ? p[7] : h[5];
        bm[2].x = hi ? h[2] : p[0];  bm[2].y = hi ? h[3] : p[1];
        bm[3].x = hi ? h[6] : p[4];  bm[3].y = hi ? h[7] : p[5];

        // chunk-outer, tile-inner: independent accumulators back-to-back
#pragma unroll
        for (int c = 0; c < 4; ++c)
#pragma unroll
            for (int g = 0; g < 4; ++g)
                acc[g] = __builtin_amdgcn_wmma_f32_16x16x4_f32(
                    false, a[g][c], false, bm[c], (short)0, acc[g], false, false);

        // Gate math: tiles 0..3 = i,f,g,o ; lane (m,hi) reg r holds hidden k = 8hi+r
#pragma unroll
        for (int r = 0; r < 8; ++r) {
            float iv = sigm(acc[0][r]);
            float fv = sigm(acc[1][r]);
            float gv = tanh_fast(acc[2][r]);
            float ov = sigm(acc[3][r]);
            float cv = fv * cst[r] + iv * gv;
            cst[r] = cv;
            h[r] = ov * tanh_fast(cv);
        }
    }

    // Final FC: out[b] = sigmoid(sum_k h[b][k]*fc_w[k] + fc_b)
    float partial = 0.0f;
#pragma unroll
    for (int r = 0; r < 8; ++r) partial += h[r] * fc_w[8 * hi + r];
    partial += __shfl_xor(partial, 16, 32);
    if (lane < 16) out[b0 + m] = sigm(partial + fc_b[0]);
}

extern "C" void kernel_launch(void* const* d_in, const int* in_sizes, int n_in,
                              void* d_out, int out_size, void* d_ws, size_t ws_size,
                              hipStream_t stream) {
    const int*   x     = (const int*)d_in[0];
    const float* embed = (const float*)d_in[1];
    const float* w_ih  = (const float*)d_in[2];
    const float* w_hh  = (const float*)d_in[3];
    const float* b_ih  = (const float*)d_in[4];
    const float* b_hh  = (const float*)d_in[5];
    const float* fc_w  = (const float*)d_in[6];
    const float* fc_b  = (const float*)d_in[7];
    float* out = (float*)d_out;
    float* xp  = (float*)d_ws;   // [T][64][B] fp32 = 512 MB scratch

    dim3 gridA(kB / 16, kT / 8);
    lstm_xproj<<<gridA, 32, 0, stream>>>(x, embed, w_ih, b_ih, b_hh, xp);
    lstm_scan<<<kB / 16, 32, 0, stream>>>(xp, w_hh, fc_w, fc_b, out);
}